// CustomCrossEntropyLoss_25202868093535
// MI455X (gfx1250) — compile-verified
//
#include <hip/hip_runtime.h>
#include <math.h>

// ---------------- problem constants (match reference setup_inputs) ----------
#define BB 4
#define SS 2048
#define VV 32000
#define ROWS (BB * SS)
#define TILE 4096                        // floats per TDM tile (16 KB)
#define NT   ((VV + TILE - 1) / TILE)    // 8 tiles (7 full + 3328)
#define NBUF 3                           // triple buffer: 2 TDM loads in flight
#define NTHR 256

typedef unsigned int v4u __attribute__((ext_vector_type(4)));
typedef int          v8i __attribute__((ext_vector_type(8)));
typedef int          v4i __attribute__((ext_vector_type(4)));

// 1-D TDM descriptor: copy n fp32 elements from global address gaddr to the LDS address in lds_addr.
// Group0: count=1 | lds_addr | global_addr[56:0] | type=2 ("image").
// Group1: data_size=4B, tensor_dim0=n, tile_dim0=n, tile_dim1/2 unused,
//         dim0 stride = n (irrelevant for 1-D), no pad/iterate/multicast.
__device__ __forceinline__ void tdm_load_1d(unsigned lds_addr,
                                            unsigned long long gaddr,
                                            unsigned n /* < 65536 */) {
  v4u g0;
  g0.x = 1u;                                            // count=1, user mode
  g0.y = lds_addr;                                      // LDS byte address
  g0.z = (unsigned)gaddr;                               // global addr [31:0]
  g0.w = ((unsigned)(gaddr >> 32) & 0x1FFFFFFu) | (2u << 30); // [56:32] | type=2
  v8i g1;
  g1[0] = (int)(2u << 16);                              // data_size = 4 bytes
  g1[1] = (int)((n & 0xFFFFu) << 16);                   // tensor_dim0[15:0]
  g1[2] = (int)((n >> 16) | (1u << 16));                // tensor_dim0 hi | tensor_dim1=1
  g1[3] = (int)(n << 16);                               // tile_dim0 = n
  g1[4] = 0;                                            // tile_dim1=0, tile_dim2=0
  g1[5] = (int)n;                                       // tensor_dim0_stride lo32
  g1[6] = 0;
  g1[7] = 0;
  v4i z4 = {0, 0, 0, 0};
#if __clang_major__ >= 23
  v8i z8 = {0, 0, 0, 0, 0, 0, 0, 0};
  __builtin_amdgcn_tensor_load_to_lds(g0, g1, z4, z4, z8, 0);
#else
  __builtin_amdgcn_tensor_load_to_lds(g0, g1, z4, z4, 0);
#endif
}

__device__ __forceinline__ int tile_elems(int t) {
  const int base = t * TILE;
  return (VV - base < TILE) ? (VV - base) : TILE;
}

// One block (256 threads = 8 waves) per row. Wave 0 drives a triple-buffered
// TDM pipeline (2 tiles in flight); all waves compute from LDS with a
// branch-free two-pass online softmax + argmax.
__global__ __launch_bounds__(NTHR) void ce_row_kernel(
    const float* __restrict__ pred, const int* __restrict__ target,
    float* __restrict__ nll, int* __restrict__ am) {
  __shared__ __align__(16) float buf[NBUF][TILE];
  __shared__ float red_m[NTHR];
  __shared__ float red_s[NTHR];
  __shared__ int   red_i[NTHR];

  const int row = blockIdx.x;
  const int tid = threadIdx.x;
  const bool issuer = (tid >> 5) == 0;  // wave 0 (uniform per wave)
  const unsigned long long rowBase =
      (unsigned long long)(uintptr_t)(pred + (size_t)row * VV);

  unsigned ldsb[NBUF];
#pragma unroll
  for (int i = 0; i < NBUF; ++i) ldsb[i] = (unsigned)(uintptr_t)&buf[i][0];

  // Prologue: put two tiles in flight.
  if (issuer) {
    tdm_load_1d(ldsb[0], rowBase, (unsigned)tile_elems(0));
    if (NT > 1)
      tdm_load_1d(ldsb[1], rowBase + (unsigned long long)TILE * 4u,
                  (unsigned)tile_elems(1));
  }

  float m = -INFINITY, s = 0.0f;
  int bi = 0x7FFFFFFF;

#pragma unroll
  for (int t = 0; t < NT; ++t) {
    // Ensure tile t has landed: TDM ops complete in order per wave, and at
    // most one load newer than t is outstanding here.
    if (issuer) {
      if (t + 1 < NT) __builtin_amdgcn_s_wait_tensorcnt(1);
      else            __builtin_amdgcn_s_wait_tensorcnt(0);
      asm volatile("" ::: "memory");
    }
    __syncthreads();  // publish tile t; fences compute(t-1) before buf reuse

    // Refill the pipeline: tile t+2 overwrites the buffer last read at t-1.
    if (issuer && t + 2 < NT)
      tdm_load_1d(ldsb[(t + 2) % NBUF],
                  rowBase + (unsigned long long)(t + 2) * TILE * 4u,
                  (unsigned)tile_elems(t + 2));

    const int base = t * TILE;
    const int n4 = tile_elems(t) >> 2;   // 1024 or 832 float4 (exact)
    const float4* p = (const float4*)&buf[t % NBUF][0];

    // Pass 1: tile max + argmax (pure v_cmp/v_cndmask, no divergence).
    float tm = -INFINITY;
    int ti = 0;
#pragma unroll 4
    for (int k = tid; k < n4; k += NTHR) {
      float4 x = p[k];
      const int idx = base + (k << 2);
      if (x.x > tm) { tm = x.x; ti = idx + 0; }
      if (x.y > tm) { tm = x.y; ti = idx + 1; }
      if (x.z > tm) { tm = x.z; ti = idx + 2; }
      if (x.w > tm) { tm = x.w; ti = idx + 3; }
    }
    const float nm = fmaxf(m, tm);
    if (tm > m) bi = ti;                 // strict >: first occurrence wins
    s *= __expf(m - nm);                 // rescale old sum (m=-inf -> 0)
    m = nm;

    // Pass 2: accumulate exp(x - m) for this tile (1 trans + 1 add / elem).
    float acc = 0.0f;
#pragma unroll 4
    for (int k = tid; k < n4; k += NTHR) {
      float4 x = p[k];
      acc += __expf(x.x - m) + __expf(x.y - m) +
             __expf(x.z - m) + __expf(x.w - m);
    }
    s += acc;
  }

  // Cross-thread combine of (max, scaled sum, argmax).
  red_m[tid] = m; red_s[tid] = s; red_i[tid] = bi;
  __syncthreads();
  for (int off = NTHR / 2; off > 0; off >>= 1) {
    if (tid < off) {
      float m1 = red_m[tid],       s1 = red_s[tid];       int i1 = red_i[tid];
      float m2 = red_m[tid + off], s2 = red_s[tid + off]; int i2 = red_i[tid + off];
      if (m2 > m1) {
        red_m[tid] = m2; red_s[tid] = s2 + s1 * __expf(m1 - m2); red_i[tid] = i2;
      } else if (m2 < m1) {
        red_s[tid] = s1 + s2 * __expf(m2 - m1);
      } else {
        red_s[tid] = s1 + s2; red_i[tid] = (i1 < i2) ? i1 : i2;
      }
    }
    __syncthreads();
  }

  if (tid == 0) {
    const int tgt = target[row];
    const float logit_t = pred[(size_t)row * VV + tgt];
    nll[row] = (red_m[0] + logf(red_s[0])) - logit_t;   // M + log S - x_t
    am[row]  = red_i[0];
  }
}

// Single-block epilogue: CE mean over valid tokens, first-EOS position and
// NEXT_LINE count for argmax ids and target ids, weighted scalar.
__global__ __launch_bounds__(NTHR) void finalize_kernel(
    const float* __restrict__ nll, const int* __restrict__ am,
    const int* __restrict__ target, float* __restrict__ out) {
  __shared__ float sf[NTHR];
  __shared__ int   si[NTHR];
  __shared__ int   firsts[8];  // [0..3]=am, [4..7]=target
  __shared__ int   cnts[8];
  const int tid = threadIdx.x;

  float num = 0.0f, den = 0.0f;
  for (int r = tid; r < ROWS; r += NTHR) {
    if (target[r] != 0) { num += nll[r]; den += 1.0f; }
  }
  sf[tid] = num; __syncthreads();
  for (int off = NTHR / 2; off; off >>= 1) { if (tid < off) sf[tid] += sf[tid + off]; __syncthreads(); }
  const float ce_num = sf[0]; __syncthreads();
  sf[tid] = den; __syncthreads();
  for (int off = NTHR / 2; off; off >>= 1) { if (tid < off) sf[tid] += sf[tid + off]; __syncthreads(); }
  const float ce_den = sf[0]; __syncthreads();

  for (int which = 0; which < 2; ++which) {
    const int* ids = which ? target : am;
    for (int b = 0; b < BB; ++b) {
      const int* p = ids + b * SS;
      int loc = SS - 1;                              // forced-True last index
      for (int j = tid; j < SS; j += NTHR)
        if (p[j] == 1 && j < loc) loc = j;           // EOS_ID == 1
      si[tid] = loc; __syncthreads();
      for (int off = NTHR / 2; off; off >>= 1) {
        if (tid < off) si[tid] = min(si[tid], si[tid + off]);
        __syncthreads();
      }
      const int first = si[0]; __syncthreads();
      int c = 0;
      for (int j = tid; j <= first; j += NTHR)
        if (p[j] == 2) c++;                          // NEXT_LINE == 2
      si[tid] = c; __syncthreads();
      for (int off = NTHR / 2; off; off >>= 1) {
        if (tid < off) si[tid] += si[tid + off];
        __syncthreads();
      }
      if (tid == 0) { firsts[which * 4 + b] = first; cnts[which * 4 + b] = si[0]; }
      __syncthreads();
    }
  }

  if (tid == 0) {
    const float ce = ce_num / fmaxf(ce_den, 1.0f);
    float len_loss = 0.0f, line_loss = 0.0f;
    for (int b = 0; b < BB; ++b) {
      len_loss  += fabsf((float)(firsts[b] - firsts[4 + b]));
      line_loss += fabsf((float)(cnts[b]   - cnts[4 + b]));
    }
    len_loss  *= (1.0f / BB);
    line_loss *= (1.0f / BB);
    out[0] = 0.98f * ce + 0.01f * len_loss + 0.01f * line_loss;
  }
}

extern "C" void kernel_launch(void* const* d_in, const int* in_sizes, int n_in,
                              void* d_out, int out_size, void* d_ws, size_t ws_size,
                              hipStream_t stream) {
  const float* pred   = (const float*)d_in[0];   // [B,S,V] fp32 logits
  const int*   target = (const int*)d_in[1];     // [B,S] int32
  float* nll = (float*)d_ws;                                   // ROWS floats
  int*   am  = (int*)((char*)d_ws + ROWS * sizeof(float));     // ROWS ints

  ce_row_kernel<<<ROWS, NTHR, 0, stream>>>(pred, target, nll, am);
  finalize_kernel<<<1, NTHR, 0, stream>>>(nll, am, target, (float*)d_out);
}